// MORNLayer_54709293416908
// MI455X (gfx1250) — compile-verified
//
#include <hip/hip_runtime.h>
#include <hip/hip_bf16.h>

typedef float v2f __attribute__((ext_vector_type(2)));
typedef float v8f __attribute__((ext_vector_type(8)));

#define NNODE 50000
#define DIM   128
#define NEDGE 1000000
#define NH    8
static const size_t NDF = (size_t)NNODE * DIM;   // floats per node-feature array

// ---------- monotone float<->uint map for atomicMax on signed floats ----------
__device__ __forceinline__ unsigned fmax_map(float f) {
  unsigned u = __float_as_uint(f);
  return (u & 0x80000000u) ? ~u : (u | 0x80000000u);
}
__device__ __forceinline__ float fmax_unmap(unsigned u) {
  return (u & 0x80000000u) ? __uint_as_float(u & 0x7FFFFFFFu) : __uint_as_float(~u);
}

// ---------- core WMMA 16x16 tile GEMM over K=128, f32 in / f32 acc ----------
// A (X) layout: 16x4 f32 tile: lanes 0-15 hold K=2*(lane/16)+{0,1} ... per ISA 7.12.2
// B (W) layout: 4x16 f32 tile: VGPR v holds row K = v + 2*(lane/16), col = lane%16
__device__ __forceinline__ v8f wmma_gemm_k128(const float* __restrict__ X,
                                              const float* __restrict__ W,
                                              int row0, int col0, int lane) {
  const int mrow  = lane & 15;
  const int khalf = (lane >> 4) << 1;   // 0 or 2
  v8f acc = {};
#pragma unroll 8
  for (int k = 0; k < DIM; k += 4) {
    v2f a, b;
    const float* xr = X + (size_t)(row0 + mrow) * DIM + (k + khalf);
    a.x = xr[0];
    a.y = xr[1];
    const float* wr = W + (size_t)(k + khalf) * DIM + (col0 + mrow);
    b.x = wr[0];
    b.y = wr[DIM];
    acc = __builtin_amdgcn_wmma_f32_16x16x4_f32(false, a, false, b, (short)0, acc,
                                                false, false);
  }
  return acc;
}

// ---------- init: zero agg+s regions, set segment-max regions to mapped(-inf) ----
__global__ void hgt_init_kernel(float* zero_base, unsigned nzero,
                                unsigned* m_base, unsigned nm) {
  unsigned i = blockIdx.x * blockDim.x + threadIdx.x;
  if (i < nzero) zero_base[i] = 0.0f;
  if (i < nm)    m_base[i]    = 0x007FFFFFu;   // fmax_map(-inf)
}

// ---------- fuse relation transforms into K/V weights ----------
// job j: 0:(Wk0,att0) 1:(Wv0,msg0) 2:(Wk1,att1) 3:(Wv1,msg1) 4:(Wk0,att2) 5:(Wv0,msg2)
__global__ void hgt_fuse_weights_kernel(const float* __restrict__ Wk, const float* __restrict__ bk,
                                        const float* __restrict__ Wv, const float* __restrict__ bv,
                                        const float* __restrict__ att, const float* __restrict__ msg,
                                        float* __restrict__ Wf, float* __restrict__ bf) {
  const int j = blockIdx.y;
  const int relIdx[6] = {0, 0, 1, 1, 2, 2};
  const int typIdx[6] = {0, 0, 1, 1, 0, 0};
  const int r  = relIdx[j];
  const int tt = typIdx[j];
  const bool isK = (j & 1) == 0;
  const float* W = (isK ? Wk : Wv) + (size_t)tt * DIM * DIM;
  const float* b = (isK ? bk : bv) + (size_t)tt * DIM;
  const float* R = (isK ? att : msg) + (size_t)r * NH * 16 * 16;
  float* Wo = Wf + (size_t)j * DIM * DIM;
  float* bo = bf + (size_t)j * DIM;

  int idx = blockIdx.x * blockDim.x + threadIdx.x;
  if (idx >= DIM * DIM) return;
  int i = idx >> 7, o = idx & 127;
  int h = o >> 4, eo = o & 15;
  float acc = 0.0f;
#pragma unroll
  for (int dd = 0; dd < 16; ++dd)
    acc += W[(size_t)i * DIM + h * 16 + dd] * R[(size_t)(h * 16 + dd) * 16 + eo];
  Wo[idx] = acc;
  if (i == 0) {
    float accb = 0.0f;
#pragma unroll
    for (int dd = 0; dd < 16; ++dd)
      accb += b[h * 16 + dd] * R[(size_t)(h * 16 + dd) * 16 + eo];
    bo[o] = accb;
  }
}

// ---------- projection GEMM: out = X @ W + bias ----------
__global__ void hgt_gemm_bias_kernel(const float* __restrict__ X, const float* __restrict__ W,
                                     const float* __restrict__ bias, float* __restrict__ out,
                                     int nrows) {
  const int lane = threadIdx.x & 31;
  const int wave = threadIdx.x >> 5;
  const int row0 = blockIdx.x * 16;
  const int col0 = wave * 16;
  if (row0 >= nrows) return;
  v8f acc = wmma_gemm_k128(X, W, row0, col0, lane);
  const int col = col0 + (lane & 15);
  const int rbase = row0 + ((lane >> 4) << 3);
  const float bc = bias[col];
#pragma unroll
  for (int v = 0; v < 8; ++v)
    out[(size_t)(rbase + v) * DIM + col] = acc[v] + bc;
}

// ---------- output GEMM + skip blend: out = (X@Wa + ba)*sig(skip) + h*(1-sig) ----
__global__ void hgt_gemm_out_kernel(const float* __restrict__ X, const float* __restrict__ Wa,
                                    const float* __restrict__ ba, const float* __restrict__ skip,
                                    int skipIdx, const float* __restrict__ h,
                                    float* __restrict__ out, int nrows) {
  const int lane = threadIdx.x & 31;
  const int wave = threadIdx.x >> 5;
  const int row0 = blockIdx.x * 16;
  const int col0 = wave * 16;
  if (row0 >= nrows) return;
  v8f acc = wmma_gemm_k128(X, Wa, row0, col0, lane);
  const float alpha = 1.0f / (1.0f + __expf(-skip[skipIdx]));
  const float beta  = 1.0f - alpha;
  const int col = col0 + (lane & 15);
  const int rbase = row0 + ((lane >> 4) << 3);
  const float bc = ba[col];
#pragma unroll
  for (int v = 0; v < 8; ++v) {
    size_t off = (size_t)(rbase + v) * DIM + col;
    out[off] = (acc[v] + bc) * alpha + h[off] * beta;
  }
}

// ---------- edge pass 1: per-edge per-head logits + segment max ----------
__global__ void hgt_edge_dot_kernel(const float* __restrict__ q, const float* __restrict__ k,
                                    const int* __restrict__ src, const int* __restrict__ dst,
                                    const float* __restrict__ ew, const float* __restrict__ pri,
                                    float* __restrict__ t, unsigned* __restrict__ m, int nedges) {
  const int lane = threadIdx.x & 31;
  const int e = blockIdx.x * (blockDim.x >> 5) + (threadIdx.x >> 5);
  if (e >= nedges) return;
  const int s = src[e], d = dst[e];
  const float4 qv = *(const float4*)(q + (size_t)d * DIM + lane * 4);
  const float4 kv = *(const float4*)(k + (size_t)s * DIM + lane * 4);
  float p = qv.x * kv.x + qv.y * kv.y + qv.z * kv.z + qv.w * kv.w;
  p += __shfl_xor(p, 1, 32);
  p += __shfl_xor(p, 2, 32);           // 4 lanes per head reduced
  const int hh = lane >> 2;
  const float tv = p * ew[e] * pri[hh] * 0.25f;   // 1/sqrt(16)
  if ((lane & 3) == 0) {
    t[(size_t)e * NH + hh] = tv;
    atomicMax(&m[(size_t)d * NH + hh], fmax_map(tv));
  }
}

// ---------- edge pass 2: unnormalized exp-weighted aggregate + denominator ----
__global__ void hgt_edge_agg_kernel(const float* __restrict__ vsrc, const int* __restrict__ src,
                                    const int* __restrict__ dst, const float* __restrict__ t,
                                    const unsigned* __restrict__ m, float* __restrict__ sden,
                                    float* __restrict__ agg, int nedges) {
  const int lane = threadIdx.x & 31;
  const int e = blockIdx.x * (blockDim.x >> 5) + (threadIdx.x >> 5);
  if (e >= nedges) return;
  const int s = src[e], d = dst[e];
  const int hh = lane >> 2;
  const float tv = t[(size_t)e * NH + hh];
  const float mv = fmax_unmap(m[(size_t)d * NH + hh]);
  const float ev = __expf(tv - mv);
  if ((lane & 3) == 0) atomicAdd(&sden[(size_t)d * NH + hh], ev);
  const float4 vv = *(const float4*)(vsrc + (size_t)s * DIM + lane * 4);
  float* ap = agg + (size_t)d * DIM + lane * 4;
  atomicAdd(ap + 0, ev * vv.x);
  atomicAdd(ap + 1, ev * vv.y);
  atomicAdd(ap + 2, ev * vv.z);
  atomicAdd(ap + 3, ev * vv.w);
}

// ---------- normalize + cross-reduce ----------
__global__ void hgt_combine_A_kernel(const float* __restrict__ agg1, const float* __restrict__ s1,
                                     const float* __restrict__ agg2, const float* __restrict__ s2,
                                     float* __restrict__ tA) {
  int idx = blockIdx.x * blockDim.x + threadIdx.x;
  if (idx >= (int)NDF) return;
  int n = idx >> 7, c = idx & 127, hh = c >> 4;
  float d1 = s1[(size_t)n * NH + hh], d2 = s2[(size_t)n * NH + hh];
  float x1 = d1 > 0.0f ? agg1[idx] / d1 : 0.0f;
  float x2 = d2 > 0.0f ? agg2[idx] / d2 : 0.0f;
  tA[idx] = 0.5f * (x1 + x2);
}
__global__ void hgt_combine_B_kernel(const float* __restrict__ agg, const float* __restrict__ s,
                                     float* __restrict__ tB) {
  int idx = blockIdx.x * blockDim.x + threadIdx.x;
  if (idx >= (int)NDF) return;
  int n = idx >> 7, c = idx & 127, hh = c >> 4;
  float d = s[(size_t)n * NH + hh];
  tB[idx] = d > 0.0f ? agg[idx] / d : 0.0f;
}

extern "C" void kernel_launch(void* const* d_in, const int* in_sizes, int n_in,
                              void* d_out, int out_size, void* d_ws, size_t ws_size,
                              hipStream_t stream) {
  const float* h_A   = (const float*)d_in[0];
  const float* h_B   = (const float*)d_in[1];
  const float* Wk    = (const float*)d_in[2];
  const float* bk    = (const float*)d_in[3];
  const float* Wq    = (const float*)d_in[4];
  const float* bq    = (const float*)d_in[5];
  const float* Wv    = (const float*)d_in[6];
  const float* bv    = (const float*)d_in[7];
  const float* Wa    = (const float*)d_in[8];
  const float* ba    = (const float*)d_in[9];
  const float* att   = (const float*)d_in[10];
  const float* msg   = (const float*)d_in[11];
  const float* pri   = (const float*)d_in[12];
  const float* skip  = (const float*)d_in[13];
  const int*   src0  = (const int*)d_in[14];
  const int*   dst0  = (const int*)d_in[15];
  const float* ew0   = (const float*)d_in[16];
  const int*   src1  = (const int*)d_in[17];
  const int*   dst1  = (const int*)d_in[18];
  const float* ew1   = (const float*)d_in[19];
  const int*   src2  = (const int*)d_in[20];
  const int*   dst2  = (const int*)d_in[21];
  const float* ew2   = (const float*)d_in[22];

  // ---- workspace layout (floats) ----
  float* ws = (float*)d_ws;
  size_t off = 0;
  float* qA   = ws + off; off += NDF;
  float* qB   = ws + off; off += NDF;
  float* k0   = ws + off; off += NDF;
  float* k1   = ws + off; off += NDF;
  float* k2   = ws + off; off += NDF;
  float* v0   = ws + off; off += NDF;
  float* v1   = ws + off; off += NDF;
  float* v2   = ws + off; off += NDF;
  float* Wf   = ws + off; off += (size_t)6 * DIM * DIM;
  float* bf   = ws + off; off += (size_t)6 * DIM;
  float* tbuf = ws + off; off += (size_t)NEDGE * NH;      // reused across relations
  // zero-init region starts here:
  float* aggB  = ws + off; off += NDF;
  float* aggA1 = ws + off; off += NDF;
  float* aggA2 = ws + off; off += NDF;
  float* sB    = ws + off; off += (size_t)NNODE * NH;
  float* sA1   = ws + off; off += (size_t)NNODE * NH;
  float* sA2   = ws + off; off += (size_t)NNODE * NH;
  const size_t nzero = (size_t)(3 * NDF + 3 * (size_t)NNODE * NH);
  unsigned* mB  = (unsigned*)(ws + off); off += (size_t)NNODE * NH;
  unsigned* mA1 = (unsigned*)(ws + off); off += (size_t)NNODE * NH;
  unsigned* mA2 = (unsigned*)(ws + off); off += (size_t)NNODE * NH;
  const size_t nm = (size_t)3 * NNODE * NH;
  float* tA = ws + off; off += NDF;
  float* tB = ws + off; off += NDF;

  float* outA = (float*)d_out;
  float* outB = (float*)d_out + NDF;

  const int THREADS = 256;

  // 1) init accumulators
  {
    unsigned maxn = (unsigned)(nzero > nm ? nzero : nm);
    hgt_init_kernel<<<(maxn + THREADS - 1) / THREADS, THREADS, 0, stream>>>(
        aggB, (unsigned)nzero, mB, (unsigned)nm);
  }

  // 2) fuse relation transforms into K/V weights
  {
    dim3 grid((DIM * DIM + THREADS - 1) / THREADS, 6);
    hgt_fuse_weights_kernel<<<grid, THREADS, 0, stream>>>(Wk, bk, Wv, bv, att, msg, Wf, bf);
  }

  // 3) projections (WMMA GEMMs), 3125 row-strips x 8 col-tile waves
  const int GRID_GEMM = NNODE / 16;
  hgt_gemm_bias_kernel<<<GRID_GEMM, THREADS, 0, stream>>>(h_A, Wq + 0 * DIM * DIM, bq + 0 * DIM, qA, NNODE);
  hgt_gemm_bias_kernel<<<GRID_GEMM, THREADS, 0, stream>>>(h_B, Wq + 1 * DIM * DIM, bq + 1 * DIM, qB, NNODE);
  hgt_gemm_bias_kernel<<<GRID_GEMM, THREADS, 0, stream>>>(h_A, Wf + 0 * DIM * DIM, bf + 0 * DIM, k0, NNODE);
  hgt_gemm_bias_kernel<<<GRID_GEMM, THREADS, 0, stream>>>(h_A, Wf + 1 * DIM * DIM, bf + 1 * DIM, v0, NNODE);
  hgt_gemm_bias_kernel<<<GRID_GEMM, THREADS, 0, stream>>>(h_B, Wf + 2 * DIM * DIM, bf + 2 * DIM, k1, NNODE);
  hgt_gemm_bias_kernel<<<GRID_GEMM, THREADS, 0, stream>>>(h_B, Wf + 3 * DIM * DIM, bf + 3 * DIM, v1, NNODE);
  hgt_gemm_bias_kernel<<<GRID_GEMM, THREADS, 0, stream>>>(h_A, Wf + 4 * DIM * DIM, bf + 4 * DIM, k2, NNODE);
  hgt_gemm_bias_kernel<<<GRID_GEMM, THREADS, 0, stream>>>(h_A, Wf + 5 * DIM * DIM, bf + 5 * DIM, v2, NNODE);

  // 4) edge phase: one 32-lane wave per edge, 8 edges per 256-thread block
  const int GRID_EDGE = (NEDGE + 7) / 8;
  // relation 0: A -> B  (q from B, k/v from A via fused rel-0 weights)
  hgt_edge_dot_kernel<<<GRID_EDGE, THREADS, 0, stream>>>(qB, k0, src0, dst0, ew0, pri + 0, tbuf, mB, NEDGE);
  hgt_edge_agg_kernel<<<GRID_EDGE, THREADS, 0, stream>>>(v0, src0, dst0, tbuf, mB, sB, aggB, NEDGE);
  // relation 1: B -> A
  hgt_edge_dot_kernel<<<GRID_EDGE, THREADS, 0, stream>>>(qA, k1, src1, dst1, ew1, pri + 8, tbuf, mA1, NEDGE);
  hgt_edge_agg_kernel<<<GRID_EDGE, THREADS, 0, stream>>>(v1, src1, dst1, tbuf, mA1, sA1, aggA1, NEDGE);
  // relation 2: A -> A
  hgt_edge_dot_kernel<<<GRID_EDGE, THREADS, 0, stream>>>(qA, k2, src2, dst2, ew2, pri + 16, tbuf, mA2, NEDGE);
  hgt_edge_agg_kernel<<<GRID_EDGE, THREADS, 0, stream>>>(v2, src2, dst2, tbuf, mA2, sA2, aggA2, NEDGE);

  // 5) normalize + mean cross-reduce
  {
    int n = (int)NDF;
    hgt_combine_A_kernel<<<(n + THREADS - 1) / THREADS, THREADS, 0, stream>>>(aggA1, sA1, aggA2, sA2, tA);
    hgt_combine_B_kernel<<<(n + THREADS - 1) / THREADS, THREADS, 0, stream>>>(aggB, sB, tB);
  }

  // 6) output GEMMs with sigmoid-skip blend
  hgt_gemm_out_kernel<<<GRID_GEMM, THREADS, 0, stream>>>(tA, Wa + 0 * DIM * DIM, ba + 0 * DIM, skip, 0, h_A, outA, NNODE);
  hgt_gemm_out_kernel<<<GRID_GEMM, THREADS, 0, stream>>>(tB, Wa + 1 * DIM * DIM, ba + 1 * DIM, skip, 1, h_B, outB, NNODE);
}